// Model_41807211659983
// MI455X (gfx1250) — compile-verified
//
#include <hip/hip_runtime.h>
#include <stdint.h>

// Problem constants (from reference setup_inputs)
#define N_B   4
#define H_C   136
#define W_C   240
#define HW_C  (H_C * W_C)
#define DSF   8
#define H_O   (H_C * DSF)      // 1088
#define W_O   (W_C * DSF)      // 1920
#define TX    32               // coarse x per block
#define NTX   ((W_C + TX - 1) / TX)  // 8
#define LDS_ROW    36          // 32 dwords data + 4 dwords TDM pad (bank-conflict-free)
#define CHUNK_ROWS 72          // 9 taps * 8 sx planes
#define CHUNK_DW   (CHUNK_ROWS * LDS_ROW)

typedef unsigned int v4u __attribute__((ext_vector_type(4)));
typedef int          v8i __attribute__((ext_vector_type(8)));
typedef int          v4i __attribute__((ext_vector_type(4)));

#if defined(__HIP_DEVICE_COMPILE__)
#  if __has_builtin(__builtin_amdgcn_tensor_load_to_lds) && __has_builtin(__builtin_amdgcn_s_wait_tensorcnt)
#    define HAVE_TDM 1
#  else
#    define HAVE_TDM 0
#  endif
#else
#  define HAVE_TDM 0
#endif

#if HAVE_TDM
// Issue one 3D TDM chunk: tile (x=32, sx=8, k=9) of the mask, padded into LDS
// rows of 36 dwords. gaddr points at mask[n, sy*8 plane, y, x0].
__device__ __attribute__((always_inline)) void tdm_issue(const float* gaddr,
                                                         unsigned lds_byte_off,
                                                         int tensor_dim0) {
  const unsigned long long ga = (unsigned long long)(uintptr_t)gaddr;
  v4u g0;
  g0[0] = 1u;                                    // count=1, user descriptor
  g0[1] = lds_byte_off;                          // lds_addr
  g0[2] = (unsigned)(ga & 0xFFFFFFFFu);          // global_addr[31:0]
  g0[3] = (unsigned)((ga >> 32) & 0x1FFFFFFu)    // global_addr[56:32]
        | (2u << 30);                            // type = 2 ("image")

  const unsigned s0 = (unsigned)HW_C;            // tensor_dim0_stride (elements)
  const unsigned long long s1 = 64ull * HW_C;    // tensor_dim1_stride (elements)
  const unsigned td0 = (unsigned)tensor_dim0;

  v8i g1;
  g1[0] = (int)((2u << 16)       // data_size = 4 bytes
              | (1u << 20)       // pad_enable
              | (4u << 22)       // pad_interval: 32 dwords
              | (3u << 25));     // pad_amount: 4 dwords
  g1[1] = (int)(td0 << 16);                            // tensor_dim0[15:0]
  g1[2] = (int)((td0 >> 16) | (8u << 16));             // tensor_dim0 hi | tensor_dim1=8
  g1[3] = (int)(32u << 16);                            // tensor_dim1 hi=0 | tile_dim0=32
  g1[4] = (int)(8u | (9u << 16));                      // tile_dim1=8 | tile_dim2=9
  g1[5] = (int)s0;                                     // tensor_dim0_stride[31:0]
  g1[6] = (int)(((unsigned)(s1 & 0xFFFFull)) << 16);   // s0 hi=0 | s1[15:0]
  g1[7] = (int)(s1 >> 16);                             // s1[47:16]

  v4i g2;
  g2[0] = 9;   // tensor_dim2 (k extent for OOB)
  g2[1] = 0;   // tensor_dim3
  g2[2] = 0;   // tensor_dim2_stride lo (tile_dim3 unused)
  g2[3] = 0;   // stride hi | tile_dim3=0
  v4i g3; g3[0] = 0; g3[1] = 0; g3[2] = 0; g3[3] = 0;

#if __clang_major__ >= 23
  v8i g4; g4[0]=0; g4[1]=0; g4[2]=0; g4[3]=0; g4[4]=0; g4[5]=0; g4[6]=0; g4[7]=0;
  __builtin_amdgcn_tensor_load_to_lds(g0, g1, g2, g3, g4, 0);
#else
  __builtin_amdgcn_tensor_load_to_lds(g0, g1, g2, g3, 0);
#endif
}
#endif  // HAVE_TDM

__global__ __launch_bounds__(256) void upflow_kernel(const float* __restrict__ flow,
                                                     const float* __restrict__ mask,
                                                     float* __restrict__ out) {
  __shared__ float smem[2 * CHUNK_DW];   // double-buffered mask chunks (20.7 KB)

  const int tid = threadIdx.x;
  int b = blockIdx.x;
  const int tx = b % NTX;  b /= NTX;
  const int y  = b % H_C;
  const int nb = b / H_C;
  const int x0 = tx * TX;

  // tid -> (xi, sx) so fine-X = x0*8 + tid (fully coalesced stores)
  const int xi = tid >> 3;
  const int sx = tid & 7;
  const int x  = x0 + xi;
  const bool valid = (x < W_C);

  // ---- Flow 3x3 patch, both channels, zero-padded borders (L2-resident) ----
  float f0[9], f1[9];
  const float* fl0 = flow + (size_t)(nb * 2 + 0) * HW_C;
  const float* fl1 = fl0 + HW_C;
#pragma unroll
  for (int dy = 0; dy < 3; ++dy) {
#pragma unroll
    for (int dx = 0; dx < 3; ++dx) {
      const int k  = dy * 3 + dx;
      const int yy = y + dy - 1;
      const int xx = x + dx - 1;
      const bool ok = ((unsigned)yy < (unsigned)H_C) && ((unsigned)xx < (unsigned)W_C);
      const int off = yy * W_C + xx;
      f0[k] = ok ? fl0[off] : 0.0f;
      f1[k] = ok ? fl1[off] : 0.0f;
    }
  }

  const float* mbase = mask + (size_t)nb * (9 * DSF * DSF) * HW_C + (size_t)y * W_C + x0;
  const int tdim0 = W_C - x0;  // TDM OOB-zeroes x beyond this (partial last tile)
  const unsigned lds0 = (unsigned)(uintptr_t)(void*)&smem[0];

#if HAVE_TDM
  if (tid < 32) {  // wave 0 owns the TDM pipeline
    tdm_issue(mbase, lds0, tdim0);  // chunk sy=0 -> buffer 0
  }
#endif

#pragma unroll
  for (int sy = 0; sy < DSF; ++sy) {
    const float* buf = &smem[(sy & 1) * CHUNK_DW];

#if HAVE_TDM
    if (tid < 32) {
      if (sy + 1 < DSF) {
        // buffer (sy+1)&1 was released at the trailing barrier of iter sy-1
        tdm_issue(mbase + (size_t)(sy + 1) * DSF * HW_C,
                  lds0 + (unsigned)(((sy + 1) & 1) * CHUNK_DW * 4), tdim0);
        __builtin_amdgcn_s_wait_tensorcnt(1);  // chunk sy done (TDM in-order)
      } else {
        __builtin_amdgcn_s_wait_tensorcnt(0);
      }
    }
    __syncthreads();  // publish chunk sy to all waves
#else
    // Cooperative fallback: 72 rows * 32 floats = 9 elements/thread
    __syncthreads();  // prior readers of this buffer done
    float* wbuf = &smem[(sy & 1) * CHUNK_DW];
#pragma unroll
    for (int j = 0; j < 9; ++j) {
      const int idx = j * 256 + tid;
      const int pl  = idx >> 5;         // 0..71 : (k*8 + sx)
      const int xc  = idx & 31;
      const int k   = pl >> 3;
      const int sxx = pl & 7;
      const int xg  = x0 + xc;
      float v = 0.0f;
      if (xg < W_C) v = mbase[(size_t)(k * 64 + sy * 8 + sxx) * HW_C + xc];
      wbuf[pl * LDS_ROW + xc] = v;
    }
    __syncthreads();
#endif

    // ---- softmax over 9 taps + convex combination, both channels ----
    float m[9];
#pragma unroll
    for (int k = 0; k < 9; ++k) m[k] = buf[(k * 8 + sx) * LDS_ROW + xi];
    float mx = m[0];
#pragma unroll
    for (int k = 1; k < 9; ++k) mx = fmaxf(mx, m[k]);
    float s = 0.0f, a0 = 0.0f, a1 = 0.0f;
#pragma unroll
    for (int k = 0; k < 9; ++k) {
      const float e = __expf(m[k] - mx);
      s += e;
      a0 = fmaf(e, f0[k], a0);
      a1 = fmaf(e, f1[k], a1);
    }

    if (valid) {
      const float scale = 8.0f / s;  // ds * flow folded into normalization
      const int Y = y * DSF + sy;
      const int X = x0 * DSF + tid;  // = (x0+xi)*8 + sx
      float* o = out + ((size_t)(nb * 2) * H_O + Y) * W_O + X;
      o[0] = a0 * scale;
      o[(size_t)H_O * W_O] = a1 * scale;
    }
    __syncthreads();  // release buffer (sy&1) for reuse by TDM issue in iter sy+1
  }
}

extern "C" void kernel_launch(void* const* d_in, const int* in_sizes, int n_in,
                              void* d_out, int out_size, void* d_ws, size_t ws_size,
                              hipStream_t stream) {
  const float* flow = (const float*)d_in[0];  // (4, 2, 136, 240) f32
  const float* mask = (const float*)d_in[1];  // (4, 576, 136, 240) f32
  float* out = (float*)d_out;                 // (4, 2, 1088, 1920) f32
  (void)in_sizes; (void)n_in; (void)out_size; (void)d_ws; (void)ws_size;

  dim3 grid(N_B * H_C * NTX);  // 4352 blocks
  dim3 block(256);             // 8 wave32
  upflow_kernel<<<grid, block, 0, stream>>>(flow, mask, out);
}